// GcnForBatching_68461778698659
// MI455X (gfx1250) — compile-verified
//
#include <hip/hip_runtime.h>
#include <hip/hip_bf16.h>

// ---------------------------------------------------------------------------
// GCN forward for MI455X (gfx1250, wave32).
// Inputs (setup_inputs order):
//   0: x        [N_NODES, 11]  f32
//   1: edge_idx [2, N_EDGES]   i32 (JAX default x64-off => int32)
//   2: y        [128, 19]      f32 (unused)
//   3: batch    [N_NODES]      i32 (sorted)
//   4: W_conv   [11, 64]       f32
//   5: b_conv   [64]           f32
//   6: W_lin    [64, 19]       f32
//   7: b_lin    [19]           f32
// Output: [128, 19] f32
// ---------------------------------------------------------------------------

#define N_NODES   1000000
#define N_EDGES   2000000
#define N_FEAT    11
#define HIDDEN    64
#define N_CLASSES 19
#define BATCH     128

typedef __attribute__((ext_vector_type(2))) float v2f;
typedef __attribute__((ext_vector_type(8))) float v8f;

// ---------------------------------------------------------------------------
__global__ void zero_kernel(float* __restrict__ p, long n) {
    long i = (long)blockIdx.x * blockDim.x + threadIdx.x;
    if (i < n) p[i] = 0.0f;
}

// deg[col[e]] += 1
__global__ void deg_kernel(const int* __restrict__ col, float* __restrict__ deg) {
    long e = (long)blockIdx.x * blockDim.x + threadIdx.x;
    if (e >= N_EDGES) return;
    atomicAdd(&deg[col[e]], 1.0f);
}

// deg -> dinv in place
__global__ void dinv_kernel(float* __restrict__ deg) {
    long i = (long)blockIdx.x * blockDim.x + threadIdx.x;
    if (i >= N_NODES) return;
    float d = deg[i];
    deg[i] = (d > 0.0f) ? rsqrtf(d) : 0.0f;
}

// ---------------------------------------------------------------------------
// xw = x @ W_conv via V_WMMA_F32_16X16X4_F32.
// One wave32 computes a 16(node) x 64(hidden) tile with 4 v8f accumulators.
// K = 11 padded to 12 -> 3 K-steps of 4.
// A layout (16x4 f32): lanes 0-15 hold {K=k0,k0+1}, lanes 16-31 hold {k0+2,k0+3}.
// B layout (4x16 f32): symmetric; N = lane&15.
// C/D layout: VGPR r -> M = r (lanes 0-15) / r+8 (lanes 16-31), N = lane&15.
// Zero-padding of K=11 is done with clamped (always in-bounds) loads followed
// by register selects, so EXEC is never modified inside the WMMA loop.
// ---------------------------------------------------------------------------
__global__ void xw_wmma_kernel(const float* __restrict__ x,
                               const float* __restrict__ Wc,
                               float* __restrict__ xw) {
    const int lane = threadIdx.x & 31;
    const int wave = threadIdx.x >> 5;
    const int NT   = N_NODES / 16;                       // 62500 tiles
    const long tile = (long)blockIdx.x * (blockDim.x >> 5) + wave;
    if (tile >= NT) return;                              // wave-uniform: EXEC stays full

    const int m0   = (int)tile * 16;
    const int half = lane >> 4;                          // 0 or 1
    const int l15  = lane & 15;

    v8f acc[4] = {v8f{}, v8f{}, v8f{}, v8f{}};

#pragma unroll
    for (int ks = 0; ks < 3; ++ks) {
        const int k0 = ks * 4 + half * 2;                // this lane's K base
        const int kc0 = (k0     < N_FEAT) ? k0     : (N_FEAT - 1);  // clamped, in-bounds
        const int kc1 = (k0 + 1 < N_FEAT) ? k0 + 1 : (N_FEAT - 1);
        const bool v0 = (k0     < N_FEAT);
        const bool v1 = (k0 + 1 < N_FEAT);

        // --- A fragment: x[m0+l15][k0 .. k0+1], zero-padded past N_FEAT
        const long mrow = (long)(m0 + l15) * N_FEAT;
        const float ax = x[mrow + kc0];
        const float ay = x[mrow + kc1];
        v2f a;
        a.x = v0 ? ax : 0.0f;
        a.y = v1 ? ay : 0.0f;

#pragma unroll
        for (int nt = 0; nt < 4; ++nt) {
            const int n = nt * 16 + l15;
            const float bx = Wc[kc0 * HIDDEN + n];
            const float by = Wc[kc1 * HIDDEN + n];
            v2f b;
            b.x = v0 ? bx : 0.0f;
            b.y = v1 ? by : 0.0f;
            acc[nt] = __builtin_amdgcn_wmma_f32_16x16x4_f32(
                /*neg_a=*/false, a, /*neg_b=*/false, b,
                /*c_mod=*/(short)0, acc[nt],
                /*reuse_a=*/false, /*reuse_b=*/false);
        }
    }

    // store D tiles (lanes with consecutive l15 write consecutive n: coalesced)
#pragma unroll
    for (int nt = 0; nt < 4; ++nt) {
        const int n = nt * 16 + l15;
#pragma unroll
        for (int r = 0; r < 8; ++r) {
            const int m = m0 + r + half * 8;
            xw[(long)m * HIDDEN + n] = acc[nt][r];
        }
    }
}

// ---------------------------------------------------------------------------
// agg[col[e]] += xw[row[e]] * (dinv[row]*dinv[col]); one thread per (edge, h)
// ---------------------------------------------------------------------------
__global__ void edge_msg_kernel(const int* __restrict__ row,
                                const int* __restrict__ col,
                                const float* __restrict__ dinv,
                                const float* __restrict__ xw,
                                float* __restrict__ agg) {
    long gid = (long)blockIdx.x * blockDim.x + threadIdx.x;
    long e = gid >> 6;
    int  h = (int)(gid & 63);
    if (e >= N_EDGES) return;
    int s = row[e];
    int t = col[e];
    float nrm = dinv[s] * dinv[t];
    if (nrm != 0.0f) {
        atomicAdd(&agg[(long)t * HIDDEN + h], xw[(long)s * HIDDEN + h] * nrm);
    }
}

// ---------------------------------------------------------------------------
// h = relu(agg + b_conv); segmented sum into pooled[batch] + counts.
// blockDim = 64 (thread = hidden dim); each block walks 64 sorted nodes and
// flushes atomics only on batch-id change.
// ---------------------------------------------------------------------------
__global__ void pool_kernel(const float* __restrict__ agg,
                            const float* __restrict__ bconv,
                            const int* __restrict__ batch,
                            float* __restrict__ pooled,
                            float* __restrict__ counts) {
    const int  t    = threadIdx.x;                   // 0..63
    const long base = (long)blockIdx.x * 64;
    const float bias = bconv[t];

    int   cur   = batch[base];
    float local = 0.0f;
    float cnt   = 0.0f;
    for (int i = 0; i < 64; ++i) {
        const long n = base + i;
        const int  b = batch[n];
        if (b != cur) {
            atomicAdd(&pooled[(long)cur * HIDDEN + t], local);
            if (t == 0) atomicAdd(&counts[cur], cnt);
            local = 0.0f; cnt = 0.0f; cur = b;
        }
        float v = agg[n * HIDDEN + t] + bias;
        local += (v > 0.0f) ? v : 0.0f;
        cnt   += 1.0f;
    }
    atomicAdd(&pooled[(long)cur * HIDDEN + t], local);
    if (t == 0) atomicAdd(&counts[cur], cnt);
}

// ---------------------------------------------------------------------------
// out = (pooled / max(counts,1)) @ W_lin + b_lin   [128 x 19]
// ---------------------------------------------------------------------------
__global__ void lin_kernel(const float* __restrict__ pooled,
                           const float* __restrict__ counts,
                           const float* __restrict__ Wl,
                           const float* __restrict__ bl,
                           float* __restrict__ out) {
    int gid = blockIdx.x * blockDim.x + threadIdx.x;
    if (gid >= BATCH * N_CLASSES) return;
    int b = gid / N_CLASSES;
    int c = gid % N_CLASSES;
    float cnt = counts[b];
    float inv = 1.0f / ((cnt > 1.0f) ? cnt : 1.0f);
    float s = 0.0f;
#pragma unroll 8
    for (int h = 0; h < HIDDEN; ++h) {
        s += (pooled[b * HIDDEN + h] * inv) * Wl[h * N_CLASSES + c];
    }
    out[gid] = s + bl[c];
}

// ---------------------------------------------------------------------------
extern "C" void kernel_launch(void* const* d_in, const int* in_sizes, int n_in,
                              void* d_out, int out_size, void* d_ws, size_t ws_size,
                              hipStream_t stream) {
    const float* x     = (const float*)d_in[0];
    const int*   eidx  = (const int*)  d_in[1];
    const int*   batch = (const int*)  d_in[3];
    const float* Wc    = (const float*)d_in[4];
    const float* bc    = (const float*)d_in[5];
    const float* Wl    = (const float*)d_in[6];
    const float* bl    = (const float*)d_in[7];
    float*       out   = (float*)d_out;

    const int* row = eidx;             // edge_index[0]
    const int* col = eidx + N_EDGES;   // edge_index[1]

    // workspace carve-up (floats)
    float* ws     = (float*)d_ws;
    float* deg    = ws;                                    // N_NODES (becomes dinv)
    float* xw     = deg + (long)N_NODES;                   // N_NODES * 64
    float* agg    = xw + (long)N_NODES * HIDDEN;           // N_NODES * 64
    float* pooled = agg + (long)N_NODES * HIDDEN;          // 128 * 64
    float* counts = pooled + (long)BATCH * HIDDEN;         // 128

    const int TB = 256;

    // 1) zero deg and (agg | pooled | counts) — contiguous tail region
    {
        long n1 = N_NODES;
        zero_kernel<<<(int)((n1 + TB - 1) / TB), TB, 0, stream>>>(deg, n1);
        long n2 = (long)N_NODES * HIDDEN + (long)BATCH * HIDDEN + BATCH;
        zero_kernel<<<(int)((n2 + TB - 1) / TB), TB, 0, stream>>>(agg, n2);
    }

    // 2) degrees over targets, then dinv in place
    deg_kernel<<<(N_EDGES + TB - 1) / TB, TB, 0, stream>>>(col, deg);
    dinv_kernel<<<(N_NODES + TB - 1) / TB, TB, 0, stream>>>(deg);

    // 3) xw = x @ W_conv via f32 WMMA (8 waves/block, 1 tile of 16 nodes/wave)
    {
        const int tiles = N_NODES / 16;        // 62500
        const int wavesPerBlock = TB / 32;     // 8
        const int grid = (tiles + wavesPerBlock - 1) / wavesPerBlock;
        xw_wmma_kernel<<<grid, TB, 0, stream>>>(x, Wc, xw);
    }

    // 4) normalized gather/scatter over edges
    {
        long total = (long)N_EDGES * HIDDEN;
        edge_msg_kernel<<<(int)((total + TB - 1) / TB), TB, 0, stream>>>(row, col, deg, xw, agg);
    }

    // 5) relu + bias + segmented mean-pool accumulation
    pool_kernel<<<N_NODES / 64, 64, 0, stream>>>(agg, bc, batch, pooled, counts);

    // 6) final linear
    {
        int total = BATCH * N_CLASSES;
        lin_kernel<<<(total + TB - 1) / TB, TB, 0, stream>>>(pooled, counts, Wl, bl, out);
    }
}